// PointBasedTransform_8933531976016
// MI455X (gfx1250) — compile-verified
//
#include <hip/hip_runtime.h>
#include <hip/hip_bf16.h>

typedef float v2f __attribute__((ext_vector_type(2)));
typedef float v8f __attribute__((ext_vector_type(8)));

#define HALF_LOG_2PI 0.91893853320467274178f

// Problem geometry (fixed by the reference): N=1024 points, 32^3 grid, 3 dims.
constexpr int NPTS  = 1024;
constexpr int JCOLS = 32 * 32 * 32 * 3;          // 98304 columns per point
constexpr int TILES_N = NPTS / 16;               // 64
constexpr int TILES_J = JCOLS / 16;              // 6144
constexpr int JT_PER_WAVE = 8;                   // column tiles per wave
constexpr int JBLOCKS = TILES_J / JT_PER_WAVE;   // 768
constexpr int BLOCK = 256;                       // 8 waves (wave32)
constexpr int TOTAL_WAVES = TILES_N * JBLOCKS;   // 49152
constexpr int NBLOCKS = TOTAL_WAVES / (BLOCK / 32); // 6144

// out[n,j] = U[n,:] . V[:,j]   (rank-8, two K=4 WMMA groups)
//   U = [ -inv2/2, inv2*p0, inv2*p1, inv2*p2, C0, C1, C2, 0 ]
//   V = [ c^2, c*e0, c*e1, c*e2, e0, e1, e2, 0 ],  e_d = (j%3==d)
__global__ __launch_bounds__(BLOCK)
void pbt_wmma_kernel(const float* __restrict__ points,
                     const float* __restrict__ confs,
                     const float* __restrict__ coords,
                     const float* __restrict__ pitchp,
                     float* __restrict__ out)
{
    const int tid  = blockIdx.x * BLOCK + (int)threadIdx.x;
    const int wave = tid >> 5;
    const int lane = (int)threadIdx.x & 31;
    const int half = lane >> 4;     // 0: lanes 0-15, 1: lanes 16-31
    const int l15  = lane & 15;

    const int ntile  = wave / JBLOCKS;
    const int jblk   = wave % JBLOCKS;
    const int jbase0 = jblk * (JT_PER_WAVE * 16);

    // ---- per-point rank coefficients, computed in-register ----
    const int   n     = ntile * 16 + l15;      // A-matrix row for this lane
    const float pitch = pitchp[0];
    const float conf  = confs[n];
    const float p0 = points[3 * n + 0];
    const float p1 = points[3 * n + 1];
    const float p2 = points[3 * n + 2];
    const float inv  = 1.0f / (conf * pitch);  // ALPHA == 1
    const float inv2 = inv * inv;
    const float U0 = -0.5f * inv2;
    const float U1 = inv2 * p0, U2 = inv2 * p1, U3 = inv2 * p2;
    const float U4 = -0.5f * inv2 * p0 * p0 - HALF_LOG_2PI;
    const float U5 = -0.5f * inv2 * p1 * p1 - HALF_LOG_2PI;
    const float U6 = -0.5f * inv2 * p2 * p2 - HALF_LOG_2PI;

    // A fragment (16x4 f32, 2 VGPRs): lanes0-15 -> K0(V0),K1(V1); lanes16-31 -> K2,K3
    v2f a_lo, a_hi;
    a_lo.x = half ? U2 : U0;   a_lo.y = half ? U3  : U1;  // basis K=0..3
    a_hi.x = half ? U6 : U4;   a_hi.y = half ? 0.f : U5;  // basis K=4..7

    for (int t = 0; t < JT_PER_WAVE; ++t) {
        const int jbase = jbase0 + t * 16;
        const int j     = jbase + l15;        // B-matrix column for this lane
        const float c   = coords[j];
        const int   d   = j % 3;
        const float e0 = (d == 0) ? 1.f : 0.f;
        const float e1 = (d == 1) ? 1.f : 0.f;
        const float e2 = (d == 2) ? 1.f : 0.f;

        // B fragment (4x16 f32, 2 VGPRs): lanes0-15 -> K0(V0),K1(V1); lanes16-31 -> K2,K3
        v2f b_lo, b_hi;
        b_lo.x = half ? c * e1 : c * c;   b_lo.y = half ? c * e2 : c * e0;
        b_hi.x = half ? e2     : e0;      b_hi.y = half ? 0.f    : e1;

        v8f acc = {};
        acc = __builtin_amdgcn_wmma_f32_16x16x4_f32(false, a_lo, false, b_lo,
                                                    (short)0, acc, false, false);
        acc = __builtin_amdgcn_wmma_f32_16x16x4_f32(false, a_hi, false, b_hi,
                                                    (short)0, acc, false, false);

        // D layout: VGPR r, lanes0-15 -> M=r, lanes16-31 -> M=8+r.
        // Row M maps to point ntile*16+M; per-row half-wave stores are 64B-contiguous.
        const size_t base_n = (size_t)(ntile * 16 + half * 8);
        float* o = out + base_n * (size_t)JCOLS + (size_t)j;
        #pragma unroll
        for (int r = 0; r < 8; ++r) {
            // 403MB write-once stream > 192MB L2 -> non-temporal stores
            __builtin_nontemporal_store(acc[r], o + (size_t)r * JCOLS);
        }
    }
}

extern "C" void kernel_launch(void* const* d_in, const int* in_sizes, int n_in,
                              void* d_out, int out_size, void* d_ws, size_t ws_size,
                              hipStream_t stream) {
    const float* points = (const float*)d_in[0];   // [1024,3]
    const float* confs  = (const float*)d_in[1];   // [1024]
    const float* coords = (const float*)d_in[2];   // [32,32,32,3] flat = 98304
    const float* pitch  = (const float*)d_in[3];   // [1]
    float* out = (float*)d_out;                    // [1024,32,32,32,3]
    (void)in_sizes; (void)n_in; (void)out_size; (void)d_ws; (void)ws_size;

    pbt_wmma_kernel<<<NBLOCKS, BLOCK, 0, stream>>>(points, confs, coords, pitch, out);
}